// SuperLoss_26499948216548
// MI455X (gfx1250) — compile-verified
//
#include <hip/hip_runtime.h>
#include <math.h>

#ifndef __has_builtin
#define __has_builtin(x) 0
#endif

// ---------- CDNA5 async global->LDS path detection ----------
#if defined(__AMDGCN__) && __has_builtin(__builtin_amdgcn_global_load_async_to_lds_b128)
#define HAVE_ASYNC_LDS 1
#else
#define HAVE_ASYNC_LDS 0
#endif

#if defined(__AMDGCN__)
#if __has_builtin(__builtin_amdgcn_s_wait_asynccnt)
#define S_WAIT_ASYNCCNT(n) __builtin_amdgcn_s_wait_asynccnt(n)
#else
#define S_WAIT_ASYNCCNT(n) asm volatile("s_wait_asynccnt %0" ::"n"(n) : "memory")
#endif
#define S_WAIT_DSCNT0() asm volatile("s_wait_dscnt 0" ::: "memory")
#define COMPILER_MEM_BARRIER() asm volatile("" ::: "memory")
#else
#define S_WAIT_ASYNCCNT(n)
#define S_WAIT_DSCNT0()
#define COMPILER_MEM_BARRIER()
#endif

// ---------- problem constants ----------
constexpr int   kC         = 32000;          // classes
constexpr int   kTPB       = 256;            // 8 waves (wave32)
constexpr int   kChunks    = kC / 4;         // 8000 float4 per row
constexpr int   kMainIters = kChunks / kTPB; // 31
constexpr int   kTail      = kChunks - kMainIters * kTPB; // 64
constexpr int   kStages    = 8;              // async pipeline depth (4KB in flight per wave)
constexpr float kLog2e     = 1.44269504088896340736f;
constexpr float kTau       = 10.373491181781864f; // log(32000)
constexpr float kE         = 2.71828182845904523536f;

#if HAVE_ASYNC_LDS
// Builtin params (confirmed by round-1 diagnostic): (v4i32 AS1*, v4i32 AS3*, Iimm, Iimm)
typedef int v4i __attribute__((ext_vector_type(4)));
typedef __attribute__((address_space(1))) v4i GV4;
typedef __attribute__((address_space(3))) v4i LV4;

__device__ __forceinline__ void async_copy_b128(const float4* gsrc, float4* lds_dst) {
  // flat->AS1 is the identity 64-bit address; flat->AS3 is addr[31:0] (LDS aperture)
  __builtin_amdgcn_global_load_async_to_lds_b128(
      (GV4*)(unsigned long long)gsrc,
      (LV4*)(unsigned int)(unsigned long long)lds_dst,
      0, 0);
}
#endif

__device__ __forceinline__ float fexp2(float x) {
#if defined(__AMDGCN__) && __has_builtin(__builtin_amdgcn_exp2f)
  return __builtin_amdgcn_exp2f(x);
#else
  return exp2f(x);
#endif
}

// Online logsumexp update for one float4 (1.25 transcendentals per element).
__device__ __forceinline__ void acc4(const float4 v, float& m, float& s) {
  float mloc = fmaxf(fmaxf(v.x, v.y), fmaxf(v.z, v.w));
  float mn   = fmaxf(m, mloc);
  s = s * fexp2((m - mn) * kLog2e)
    + fexp2((v.x - mn) * kLog2e) + fexp2((v.y - mn) * kLog2e)
    + fexp2((v.z - mn) * kLog2e) + fexp2((v.w - mn) * kLog2e);
  m = mn;
}

__device__ __forceinline__ void lse_combine(float& m, float& s, float m2, float s2) {
  float M = fmaxf(m, m2);
  s = s * fexp2((m - M) * kLog2e) + s2 * fexp2((m2 - M) * kLog2e);
  m = M;
}

__global__ __launch_bounds__(kTPB) void superloss_row_kernel(
    const float* __restrict__ logits, const int* __restrict__ targets,
    const float* __restrict__ cw, float* __restrict__ row_loss) {
  const int row = blockIdx.x;
  const int t   = threadIdx.x;
  const float4* rowp4 = reinterpret_cast<const float4*>(logits + (long long)row * kC);

  __shared__ float4 buf[kStages * kTPB];   // 32 KB: per-lane private 16B slots
  __shared__ float  red_m[kTPB / 32];
  __shared__ float  red_s[kTPB / 32];

  float m = -3.0e38f;
  float s = 0.0f;

#if HAVE_ASYNC_LDS
  // Prologue: fill the pipeline. Each lane copies its own 16B slot; ASYNCcnt is
  // per-wave and async loads complete in order, so stage i is complete once
  // outstanding <= (issued - (i+1)).
#pragma unroll
  for (int p = 0; p < kStages; ++p)
    async_copy_b128(rowp4 + p * kTPB + t, &buf[p * kTPB + t]);

#pragma unroll 1
  for (int i = 0; i < kMainIters; ++i) {
    const int rem = kMainIters - 1 - i;  // loads allowed to remain outstanding
    if (rem >= 7)      S_WAIT_ASYNCCNT(7);
    else if (rem == 6) S_WAIT_ASYNCCNT(6);
    else if (rem == 5) S_WAIT_ASYNCCNT(5);
    else if (rem == 4) S_WAIT_ASYNCCNT(4);
    else if (rem == 3) S_WAIT_ASYNCCNT(3);
    else if (rem == 2) S_WAIT_ASYNCCNT(2);
    else if (rem == 1) S_WAIT_ASYNCCNT(1);
    else               S_WAIT_ASYNCCNT(0);
    COMPILER_MEM_BARRIER();
    float4 v = buf[(i % kStages) * kTPB + t];   // ds_load_b128
    acc4(v, m, s);
    // DS reads and async LDS writes use different counters and are unordered:
    // drain DScnt before the async engine may overwrite this slot.
    S_WAIT_DSCNT0();
    if (i + kStages < kMainIters)
      async_copy_b128(rowp4 + (i + kStages) * kTPB + t,
                      &buf[(i % kStages) * kTPB + t]);
  }
#else
#pragma unroll 1
  for (int i = 0; i < kMainIters; ++i) acc4(rowp4[i * kTPB + t], m, s);
#endif

  // Tail: 64 remaining chunks, direct global loads (waves 0-1 only, uniform).
  if (t < kTail) acc4(rowp4[kMainIters * kTPB + t], m, s);

  // Wave32 reduction of (m, s).
#pragma unroll
  for (int off = 16; off > 0; off >>= 1) {
    float m2 = __shfl_xor(m, off, 32);
    float s2 = __shfl_xor(s, off, 32);
    lse_combine(m, s, m2, s2);
  }
  const int lane = t & 31, wid = t >> 5;
  if (lane == 0) { red_m[wid] = m; red_s[wid] = s; }
  __syncthreads();

  if (t == 0) {
    float M = red_m[0], S = red_s[0];
#pragma unroll
    for (int wv = 1; wv < kTPB / 32; ++wv) lse_combine(M, S, red_m[wv], red_s[wv]);

    float lse = M + logf(S);
    int   tg  = targets[row];
    float lt  = logits[(long long)row * kC + tg];
    float wgt = cw[tg];
    float l_i = (lse - lt) * wgt;                 // weighted NLL

    // sigma_fn (matches reference exactly, incl. where-guards)
    float y = 0.5f * fmaxf(-2.0f / kE, l_i - kTau);   // LAM = 1
    y = fminf(fmaxf(y, -1.0f), 10.0f);
    float w = (y < 0.0f) ? (-1.0f + sqrtf(fmaxf(2.0f * (kE * y + 1.0f), 0.0f)))
                         : log1pf(y);
#pragma unroll 1
    for (int it = 0; it < 12; ++it) {
      float ew   = expf(w);
      float f    = w * ew - y;
      float wp1  = w + 1.0f;
      float swp1 = (fabsf(wp1) < 1e-6f) ? 1e-6f : wp1;
      float den  = ew * wp1 - (w + 2.0f) * f / (2.0f * swp1);
      float sden = (fabsf(den) < 1e-30f) ? 1e-30f : den;
      w = w - f / sden;
    }
    row_loss[row] = (l_i - kTau) * expf(-w);
  }
}

__global__ __launch_bounds__(256) void mean_kernel(const float* __restrict__ in,
                                                   float* __restrict__ out, int n) {
  __shared__ float sm[256];
  float acc = 0.0f;
  for (int i = threadIdx.x; i < n; i += 256) acc += in[i];
  sm[threadIdx.x] = acc;
  __syncthreads();
#pragma unroll
  for (int stride = 128; stride > 0; stride >>= 1) {
    if ((int)threadIdx.x < stride) sm[threadIdx.x] += sm[threadIdx.x + stride];
    __syncthreads();
  }
  if (threadIdx.x == 0) out[0] = sm[0] / (float)n;
}

extern "C" void kernel_launch(void* const* d_in, const int* in_sizes, int n_in,
                              void* d_out, int out_size, void* d_ws, size_t ws_size,
                              hipStream_t stream) {
  const float* logits  = (const float*)d_in[0];
  const int*   targets = (const int*)d_in[1];
  const float* cw      = (const float*)d_in[2];
  float*       out     = (float*)d_out;
  float*       ws      = (float*)d_ws;   // n_rows floats of scratch
  const int n_rows = in_sizes[1];

  superloss_row_kernel<<<n_rows, kTPB, 0, stream>>>(logits, targets, cw, ws);
  mean_kernel<<<1, 256, 0, stream>>>(ws, out, n_rows);
}